// Clash_net_54004918780081
// MI455X (gfx1250) — compile-verified
//
#include <hip/hip_runtime.h>
#include <hip/hip_bf16.h>
#include <math.h>

// ---------------------------------------------------------------------------
// Clash-energy kernel for MI455X (gfx1250, wave32).
//
// Stage 1: streaming masked-relu reduction over 4M pairs -> 16 x NBLK f32
//          partial matrix in d_ws (rows 0..5 = classes, rows 6..15 = 0).
// Stage 2: one wave32 reduces the partial matrix with V_WMMA_F32_16X16X4_F32
//          (D += Ones(16x4) x B(4x16)), then scales by exp(weight[0]).
// ---------------------------------------------------------------------------

typedef __attribute__((ext_vector_type(2))) float v2f;
typedef __attribute__((ext_vector_type(8))) float v8f;

#define TPB  256
#define NBLK 1024
#define NROW 16   // padded class rows for the 16x16 WMMA tile

__global__ __launch_bounds__(TPB) void clash_partials_kernel(
    const float* __restrict__ coords,        // [N_ATOMS*3]
    const float* __restrict__ radii,         // [N_HASH]
    const float* __restrict__ tol6,          // [6]
    const int*   __restrict__ names,         // [N_ATOMS]
    const int*   __restrict__ pairs,         // [P*2]
    const unsigned char* __restrict__ masks, // [6*P] (bool)
    float* __restrict__ partials,            // [NROW * gridDim.x]
    int P)
{
    const int nthreads = gridDim.x * blockDim.x;
    const int tid      = blockIdx.x * blockDim.x + threadIdx.x;
    const int ngroups  = P >> 2;             // 4 pairs per group

    const float t0 = tol6[0], t1 = tol6[1], t2 = tol6[2];
    const float t3 = tol6[3], t4 = tol6[4], t5 = tol6[5];

    float a0 = 0.f, a1 = 0.f, a2 = 0.f, a3 = 0.f, a4 = 0.f, a5 = 0.f;

    for (int g = tid; g < ngroups; g += nthreads) {
        // Prefetch the next stride's streamed data (global_prefetch_b8).
        const int gn = g + nthreads;
        if (gn < ngroups) {
            __builtin_prefetch(pairs + 8 * (size_t)gn, 0, 0);
            __builtin_prefetch(masks + 4 * (size_t)gn, 0, 0);
        }

        // 4 pairs = 32B: two b128 loads.
        const int4 pa = ((const int4*)pairs)[2 * g];
        const int4 pb = ((const int4*)pairs)[2 * g + 1];

        // 6 classes x 4 packed mask bytes: six b32 loads.
        unsigned m[6];
#pragma unroll
        for (int c = 0; c < 6; ++c)
            m[c] = *(const unsigned*)(masks + (size_t)c * (size_t)P + 4 * (size_t)g);

        const int p0s[4] = { pa.x, pa.z, pb.x, pb.z };
        const int p1s[4] = { pa.y, pa.w, pb.y, pb.w };

#pragma unroll
        for (int j = 0; j < 4; ++j) {
            const int p0 = p0s[j];
            const int p1 = p1s[j];
            // coords/names/radii gathers hit L2 (arrays are ~1.2MB total).
            const float dx = coords[3 * p0 + 0] - coords[3 * p1 + 0];
            const float dy = coords[3 * p0 + 1] - coords[3 * p1 + 1];
            const float dz = coords[3 * p0 + 2] - coords[3 * p1 + 2];
            const float dist = sqrtf(dx * dx + dy * dy + dz * dz + 1e-12f);
            const float rsum = radii[names[p0]] + radii[names[p1]];
            const float x = rsum - dist;

            const int sh = 8 * j;
            a0 += (float)((m[0] >> sh) & 0xFFu) * fmaxf(x + t0, 0.0f);
            a1 += (float)((m[1] >> sh) & 0xFFu) * fmaxf(x + t1, 0.0f);
            a2 += (float)((m[2] >> sh) & 0xFFu) * fmaxf(x + t2, 0.0f);
            a3 += (float)((m[3] >> sh) & 0xFFu) * fmaxf(x + t3, 0.0f);
            a4 += (float)((m[4] >> sh) & 0xFFu) * fmaxf(x + t4, 0.0f);
            a5 += (float)((m[5] >> sh) & 0xFFu) * fmaxf(x + t5, 0.0f);
        }
    }

    // Block-level tree reduction of the 6 accumulators in LDS (6KB of 320KB).
    __shared__ float red[6][TPB];
    red[0][threadIdx.x] = a0; red[1][threadIdx.x] = a1; red[2][threadIdx.x] = a2;
    red[3][threadIdx.x] = a3; red[4][threadIdx.x] = a4; red[5][threadIdx.x] = a5;
    __syncthreads();

    for (int s = TPB / 2; s > 0; s >>= 1) {
        if (threadIdx.x < (unsigned)s) {
#pragma unroll
            for (int c = 0; c < 6; ++c)
                red[c][threadIdx.x] += red[c][threadIdx.x + s];
        }
        __syncthreads();
    }

    // Write a 16-row column (rows 6..15 zero) so stage 2's B tile is clean.
    if (threadIdx.x < NROW) {
        const float v = (threadIdx.x < 6) ? red[threadIdx.x][0] : 0.0f;
        partials[threadIdx.x * gridDim.x + blockIdx.x] = v;
    }
}

// One wave32; EXEC all-1s for the whole WMMA loop (divergence only at the
// final store). D[m,n] += sum_k 1 * B[k,n]  accumulates exact f32 class sums.
// B layout (4x16, 2 VGPRs/lane): lane l supplies column n = l&15,
// rows {0,1} for lanes 0-15 and rows {2,3} for lanes 16-31.
__global__ __launch_bounds__(32) void clash_wmma_reduce_kernel(
    const float* __restrict__ partials,   // [NROW][nblk]
    const float* __restrict__ weight,     // [1]
    float* __restrict__ out,              // [6]
    int nblk)
{
    const int lane = threadIdx.x;         // 0..31, wave32
    const int n    = lane & 15;           // B column (class row of partials)
    const int koff = (lane >> 4) << 1;    // k-rows 0/1 vs 2/3

    v2f ones; ones[0] = 1.0f; ones[1] = 1.0f;   // A = Ones(16x4)
    v8f acc = {};

    for (int base = 0; base < nblk; base += 4) {
        v2f b;
        b[0] = partials[n * nblk + base + koff];
        b[1] = partials[n * nblk + base + koff + 1];
        // v_wmma_f32_16x16x4_f32 : acc(16x16) += Ones(16x4) x B(4x16)
        acc = __builtin_amdgcn_wmma_f32_16x16x4_f32(
            /*neg_a=*/false, ones, /*neg_b=*/false, b,
            /*c_mod=*/(short)0, acc, /*reuse_a=*/false, /*reuse_b=*/false);
    }

    const float scale = expf(weight[0]);
    // acc[0] on lanes 0..15 holds D[M=0, N=lane] = total sum of class `lane`.
    if (lane < 6)
        out[lane] = acc[0] * scale;
}

extern "C" void kernel_launch(void* const* d_in, const int* in_sizes, int n_in,
                              void* d_out, int out_size, void* d_ws, size_t ws_size,
                              hipStream_t stream) {
    const float* coords = (const float*)d_in[0];
    const float* radii  = (const float*)d_in[1];
    const float* tol    = (const float*)d_in[2];
    const float* weight = (const float*)d_in[3];
    const int*   names  = (const int*)d_in[4];
    const int*   pairs  = (const int*)d_in[5];
    const unsigned char* masks = (const unsigned char*)d_in[6];

    const int P = in_sizes[5] / 2;        // 4,000,000 pairs

    float* partials = (float*)d_ws;       // NROW * NBLK floats = 64 KB

    clash_partials_kernel<<<NBLK, TPB, 0, stream>>>(
        coords, radii, tol, names, pairs, masks, partials, P);

    clash_wmma_reduce_kernel<<<1, 32, 0, stream>>>(
        partials, weight, (float*)d_out, NBLK);
}